// LightGCN_55456617726565
// MI455X (gfx1250) — compile-verified
//
#include <hip/hip_runtime.h>
#include <hip/hip_bf16.h>

// LightGCN on MI455X (gfx1250, wave32).
// - 3 propagation layers: scatter/gather with native global_atomic_add_f32;
//   entire working set (~110MB) is resident in the 192MB L2.
// - residual-accumulate / rezero / final-scale passes run per 16x16 tile on
//   the matrix pipe via V_WMMA_F32_16X16X4_F32 with an exact identity-
//   selection B matrix (bit-exact f32 adds, zero extra memory traffic).

#define NODES 100000
#define DIM   64
#define TPB   256

typedef __attribute__((ext_vector_type(2))) float v2f;
typedef __attribute__((ext_vector_type(8))) float v8f;

__device__ __forceinline__ void atom_add_f32(float* p, float v) {
    __hip_atomic_fetch_add(p, v, __ATOMIC_RELAXED, __HIP_MEMORY_SCOPE_AGENT);
}

// ---------------------------------------------------------------- zero fill
__global__ void lgcn_zero(float* __restrict__ p, int n) {
    int i = blockIdx.x * blockDim.x + threadIdx.x;
    int stride = gridDim.x * blockDim.x;
    for (; i < n; i += stride) p[i] = 0.0f;
}

// ------------------------------------------------------------ degree (dst)
__global__ void lgcn_deg(const int* __restrict__ ei, float* __restrict__ deg, int E) {
    int e = blockIdx.x * blockDim.x + threadIdx.x;
    if (e < E) {
        int d = ei[E + e];                    // edge_index[1][e]
        atom_add_f32(&deg[d], 1.0f);
    }
}

// ------------------------------------------------- dinv = rsqrt(deg) in place
__global__ void lgcn_dinv(float* __restrict__ deg, int n) {
    int i = blockIdx.x * blockDim.x + threadIdx.x;
    if (i < n) {
        float g = deg[i];
        deg[i] = (g > 0.0f) ? rsqrtf(g) : 0.0f;
    }
}

// --------------------------------------------- scatter: y[dst] += w * x[src]
// 16 lanes per edge, float4 per lane -> one coalesced 256B gather per edge,
// 64 native global_atomic_add_f32 per edge (resolved in the 192MB L2).
__global__ void lgcn_scatter(const int* __restrict__ ei,
                             const float* __restrict__ dinv,
                             const float* __restrict__ x,
                             float* __restrict__ y, int E) {
    int t = blockIdx.x * blockDim.x + threadIdx.x;
    int e = t >> 4;
    if (e >= E) return;
    int lane16 = t & 15;

    int s = ei[e];
    int d = ei[E + e];
    float w = dinv[s] * dinv[d];

    float4 v = ((const float4*)(x + (long)s * DIM))[lane16];
    float* yd = y + (long)d * DIM + lane16 * 4;
    atom_add_f32(yd + 0, v.x * w);
    atom_add_f32(yd + 1, v.y * w);
    atom_add_f32(yd + 2, v.z * w);
    atom_add_f32(yd + 3, v.w * w);
}

// --------------- WMMA tile combine: dst = (csrc + X) * scale ; zbuf = 0 ----
// One wave32 per 16x16 tile. C += X computed exactly as
//   D = sum_k  X[:,4k:4k+4] x E_k  + C,   E_k[j,n] = (n == 4k + j)
// Layouts per CDNA5 ISA 7.12.2:
//   A (16x4 f32): lane m (0-15) v0=K0 v1=K1 ; lane m+16 v0=K2 v1=K3
//   B (4x16 f32): lane n (0-15) v0=row0 v1=row1 ; lane n+16 v0=row2 v1=row3
//   C/D (16x16) : VGPR v, lanes 0-15 -> M=v, lanes 16-31 -> M=v+8, N=lane%16
template <bool ZERO>
__global__ void lgcn_combine(const float* __restrict__ xbuf,
                             const float* __restrict__ csrc,
                             float* __restrict__ dst,
                             float* __restrict__ zbuf,
                             float scale, int num_tiles) {
    int wave = (blockIdx.x * blockDim.x + threadIdx.x) >> 5;
    if (wave >= num_tiles) return;          // wave-uniform: EXEC stays all-1s
    int lane = threadIdx.x & 31;
    int n  = lane & 15;
    int hi = lane >> 4;

    const int tiles_per_row = DIM / 16;     // 4
    int tr = wave / tiles_per_row;          // tile row (nodes)
    int tc = wave % tiles_per_row;          // tile col (features)
    int base = tr * 16 * DIM + tc * 16;

    // load C accumulator tile
    v8f c;
#pragma unroll
    for (int v = 0; v < 8; ++v)
        c[v] = csrc[base + (v + 8 * hi) * DIM + n];

    // 4 chained WMMAs: c += X (exact)
#pragma unroll
    for (int k = 0; k < 4; ++k) {
        int j0 = 4 * k + 2 * hi;            // K index held by this lane half
        v2f a = *(const v2f*)(xbuf + base + n * DIM + j0);
        v2f b;
        b.x = (n == j0)     ? 1.0f : 0.0f;
        b.y = (n == j0 + 1) ? 1.0f : 0.0f;
        c = __builtin_amdgcn_wmma_f32_16x16x4_f32(
                /*neg_a=*/false, a, /*neg_b=*/false, b,
                /*c_mod=*/(short)0, c, /*reuse_a=*/false, /*reuse_b=*/false);
    }

#pragma unroll
    for (int v = 0; v < 8; ++v) {
        int off = base + (v + 8 * hi) * DIM + n;
        dst[off] = c[v] * scale;
        if constexpr (ZERO) zbuf[off] = 0.0f;
    }
}

extern "C" void kernel_launch(void* const* d_in, const int* in_sizes, int n_in,
                              void* d_out, int out_size, void* d_ws, size_t ws_size,
                              hipStream_t stream) {
    const int*   ei  = (const int*)d_in[0];      // [2, E] int32
    const float* emb = (const float*)d_in[1];    // [NODES, DIM] f32
    float*       out = (float*)d_out;            // [NODES, DIM] f32
    const int E = in_sizes[0] / 2;               // 1,280,000

    float* ws   = (float*)d_ws;
    float* dinv = ws;                            // NODES floats (deg, then rsqrt in place)
    float* bufA = ws + NODES;                    // NODES*DIM floats
    float* bufB = bufA + (long)NODES * DIM;      // NODES*DIM floats

    const int tiles      = (NODES / 16) * (DIM / 16);   // 25,000
    const int cblocks    = (tiles * 32 + TPB - 1) / TPB;
    const int sblocks    = (E * 16 + TPB - 1) / TPB;
    const int eblocks    = (E + TPB - 1) / TPB;
    const int nblocks    = (NODES + TPB - 1) / TPB;
    const int zcount     = NODES + NODES * DIM;          // deg + bufA
    const int zblocks    = 2048;

    // 1. zero deg + bufA (contiguous)
    lgcn_zero<<<zblocks, TPB, 0, stream>>>(ws, zcount);
    // 2. degree over destinations
    lgcn_deg<<<eblocks, TPB, 0, stream>>>(ei, dinv, E);
    // 3. dinv = rsqrt(deg) where deg > 0
    lgcn_dinv<<<nblocks, TPB, 0, stream>>>(dinv, NODES);
    // 4. layer 1: emb -> A
    lgcn_scatter<<<sblocks, TPB, 0, stream>>>(ei, dinv, emb, bufA, E);
    // 5. out = emb + A ; zero B      (WMMA)
    lgcn_combine<true><<<cblocks, TPB, 0, stream>>>(bufA, emb, out, bufB, 1.0f, tiles);
    // 6. layer 2: A -> B
    lgcn_scatter<<<sblocks, TPB, 0, stream>>>(ei, dinv, bufA, bufB, E);
    // 7. out += B ; zero A           (WMMA)
    lgcn_combine<true><<<cblocks, TPB, 0, stream>>>(bufB, out, out, bufA, 1.0f, tiles);
    // 8. layer 3: B -> A
    lgcn_scatter<<<sblocks, TPB, 0, stream>>>(ei, dinv, bufB, bufA, E);
    // 9. out = (out + A) * 0.25      (WMMA)
    lgcn_combine<false><<<cblocks, TPB, 0, stream>>>(bufA, out, out, nullptr, 0.25f, tiles);
}